// GaussCrossEntropyLoss0_2508260901486
// MI455X (gfx1250) — compile-verified
//
#include <hip/hip_runtime.h>
#include <hip/hip_bf16.h>
#include <math.h>

typedef float v2f __attribute__((ext_vector_type(2)));
typedef float v8f __attribute__((ext_vector_type(8)));

#define MAXB 16
#define NTHR 256
#define TILE 256
#define NBLK_STAT 2048
#define NBLK_LOSS 2048

// Order-preserving float<->int mapping so integer atomicMin/Max == float min/max.
__device__ __forceinline__ int f2o(float f) {
  int i = __float_as_int(f);
  return (i >= 0) ? i : (i ^ 0x7FFFFFFF);
}
__device__ __forceinline__ float o2f(int o) {
  int i = (o >= 0) ? o : (o ^ 0x7FFFFFFF);
  return __int_as_float(i);
}

// batch = #{j : offset[j] <= i}; nx = first offset > i (INT_MAX if none).
// offr[] lives in registers (fully unrolled).
__device__ __forceinline__ void batch_rescan(const int (&offr)[MAXB], int i,
                                             int& b, int& nx) {
  int bb = 0, nn = 0x7FFFFFFF;
#pragma unroll
  for (int j = 0; j < MAXB; ++j) {
    bool le = (offr[j] <= i);
    bb += le ? 1 : 0;
    if (!le) nn = min(nn, offr[j]);
  }
  b = bb;
  nx = nn;
}

__global__ void gcel_init_kernel(int* st) {
  int t = threadIdx.x;
  if (t < MAXB) {
    st[t]            = f2o(-INFINITY); // gmax
    st[MAXB + t]     = f2o( INFINITY); // pmin
    st[2 * MAXB + t] = f2o( INFINITY); // zmin
    st[3 * MAXB + t] = f2o(-INFINITY); // zmax
  }
}

// Pass 1: per-cloud z stats. Each block owns a contiguous chunk; each thread
// accumulates min/max in registers, flushing to LDS only on (rare) cloud
// boundary crossings; one wave shuffle-tree at the end.
__global__ void gcel_stats_kernel(const float* __restrict__ coord,
                                  const int* __restrict__ segment,
                                  const int* __restrict__ offset,
                                  int n, int B, int* st) {
  __shared__ int s_g[MAXB], s_p[MAXB], s_lo[MAXB], s_hi[MAXB];
  __shared__ int s_off[MAXB];
  int tid = threadIdx.x;
  if (tid < MAXB) {
    s_g[tid]  = f2o(-INFINITY);
    s_p[tid]  = f2o( INFINITY);
    s_lo[tid] = f2o( INFINITY);
    s_hi[tid] = f2o(-INFINITY);
    s_off[tid] = (tid < B) ? offset[tid] : 0x7FFFFFFF;
  }
  __syncthreads();

  int offr[MAXB];
#pragma unroll
  for (int j = 0; j < MAXB; ++j) offr[j] = s_off[j];

  int chunk = (n + (int)gridDim.x - 1) / (int)gridDim.x;
  int beg = (int)blockIdx.x * chunk;
  int end = min(beg + chunk, n);

  float gz = -INFINITY, pz = INFINITY, lz = INFINITY, hz = -INFINITY;
  int curb = -1, nx = 0x80000000; // force rescan on first point

  for (int i = beg + tid; i < end; i += NTHR) {
    if (i >= nx) {
      if (curb >= 0) { // flush previous cloud's register accumulators (rare)
        atomicMax(&s_g[curb],  f2o(gz));
        atomicMin(&s_p[curb],  f2o(pz));
        atomicMin(&s_lo[curb], f2o(lz));
        atomicMax(&s_hi[curb], f2o(hz));
        gz = -INFINITY; pz = INFINITY; lz = INFINITY; hz = -INFINITY;
      }
      batch_rescan(offr, i, curb, nx);
      if (curb >= B) curb = B - 1;
    }
    float z = coord[i * 3 + 2];
    int  sg = segment[i];
    lz = fminf(lz, z);
    hz = fmaxf(hz, z);
    gz = fmaxf(gz, (sg == 0) ? z : -INFINITY);
    pz = fminf(pz, (sg == 1) ? z :  INFINITY);
  }

  // Final flush: one wave tree if the whole wave ended in the same cloud.
  int bRef = __shfl(curb, 0);
  bool uni = __all(curb == bRef) && (bRef >= 0);
  if (uni) {
#pragma unroll
    for (int m = 16; m >= 1; m >>= 1) {
      gz = fmaxf(gz, __shfl_xor(gz, m));
      pz = fminf(pz, __shfl_xor(pz, m));
      lz = fminf(lz, __shfl_xor(lz, m));
      hz = fmaxf(hz, __shfl_xor(hz, m));
    }
    if ((tid & 31) == 0) {
      atomicMax(&s_g[bRef],  f2o(gz));
      atomicMin(&s_p[bRef],  f2o(pz));
      atomicMin(&s_lo[bRef], f2o(lz));
      atomicMax(&s_hi[bRef], f2o(hz));
    }
  } else if (curb >= 0) {
    atomicMax(&s_g[curb],  f2o(gz));
    atomicMin(&s_p[curb],  f2o(pz));
    atomicMin(&s_lo[curb], f2o(lz));
    atomicMax(&s_hi[curb], f2o(hz));
  }
  __syncthreads();
  if (tid < B) {
    atomicMax(&st[tid],            s_g[tid]);
    atomicMin(&st[MAXB + tid],     s_p[tid]);
    atomicMin(&st[2 * MAXB + tid], s_lo[tid]);
    atomicMax(&st[3 * MAXB + tid], s_hi[tid]);
  }
}

__global__ void gcel_mu_kernel(const int* __restrict__ st, float* __restrict__ mu, int B) {
  int t = threadIdx.x;
  if (t < B) {
    int sentG = f2o(-INFINITY), sentP = f2o(INFINITY);
    int g_o = st[t], p_o = st[MAXB + t];
    float zlo = o2f(st[2 * MAXB + t]);
    float zhi = o2f(st[3 * MAXB + t]);
    float g = (g_o == sentG) ? zlo : o2f(g_o);
    float p = (p_o == sentP) ? zhi : o2f(p_o);
    mu[t] = g + (p - g) * 0.5f;
  }
}

// Pass 2: focal CE * gaussian weight. Each block owns a contiguous tile range;
// `pred` is staged into LDS with double-buffered global_load_async_to_lds_b64
// (+ s_wait_asynccnt). Batch/mu tracked incrementally in registers. Wave
// partials reduced with V_WMMA_F32_16X16X4_F32 (A = ones -> column sums).
__global__ void gcel_loss_kernel(const float* __restrict__ pred,
                                 const float* __restrict__ coord,
                                 const int* __restrict__ segment,
                                 const int* __restrict__ offset,
                                 const float* __restrict__ mu,
                                 int n, int B, float* __restrict__ partial) {
  __shared__ float2 buf[2][TILE];
  __shared__ float s_mu[MAXB];
  __shared__ int s_off[MAXB];
  __shared__ float s_wave[NTHR / 32];

  int tid = threadIdx.x;
  if (tid < MAXB) {
    s_mu[tid]  = (tid < B) ? mu[tid] : 0.f;
    s_off[tid] = (tid < B) ? offset[tid] : 0x7FFFFFFF;
  }
  __syncthreads();

  int offr[MAXB];
#pragma unroll
  for (int j = 0; j < MAXB; ++j) offr[j] = s_off[j];

  int ntiles = (n + TILE - 1) / TILE;
  int tpb    = (ntiles + (int)gridDim.x - 1) / (int)gridDim.x;
  int tbeg   = (int)blockIdx.x * tpb;
  int tend   = min(tbeg + tpb, ntiles);

  float acc = 0.f;
  int cur = 0;
  int curb = -1, nx = 0x80000000; // force first rescan
  float muv = 0.f;

  if (tbeg < tend) { // prefetch first tile into buf[0]
    int p = tbeg * TILE + tid;
    if (p < n) {
      unsigned lds = (unsigned)(uintptr_t)(void*)&buf[0][tid];
      unsigned go  = (unsigned)p * 8u;
      asm volatile("global_load_async_to_lds_b64 %0, %1, %2"
                   :: "v"(lds), "v"(go), "s"(pred) : "memory");
    }
  }

  for (int t = tbeg; t < tend; ++t) {
    asm volatile("s_wait_asynccnt 0" ::: "memory");
    __syncthreads(); // staged tile visible to all waves

    if (t + 1 < tend) { // prefetch next (contiguous) tile into other buffer
      int p = (t + 1) * TILE + tid;
      if (p < n) {
        unsigned lds = (unsigned)(uintptr_t)(void*)&buf[cur ^ 1][tid];
        unsigned go  = (unsigned)p * 8u;
        asm volatile("global_load_async_to_lds_b64 %0, %1, %2"
                     :: "v"(lds), "v"(go), "s"(pred) : "memory");
      }
    }

    int i = t * TILE + tid; // increases by TILE per iteration -> monotone
    if (i < n) {
      if (i >= nx) {
        batch_rescan(offr, i, curb, nx);
        if (curb >= B) curb = B - 1;
        muv = s_mu[curb];
      }
      float2 pr = buf[cur][tid];
      int   sg  = segment[i];
      float z   = coord[i * 3 + 2];

      float p0 = pr.x, p1 = pr.y;
      float mxv = fmaxf(p0, p1);
      float mnv = fminf(p0, p1);
      float lse = mxv + log1pf(expf(mnv - mxv)); // logsumexp, C==2
      float lt  = (sg ? p1 : p0) - lse;          // log p_t
      float pt  = expf(lt);
      float om  = 1.f - pt;
      float l   = -lt * om * om;                 // CE * focal (alpha=1, gamma=2)

      float d  = z - muv;
      float dd = d * d;
      float wl = expf(dd * -50.0f);              // 1/(2*0.1^2)
      float wr = expf(dd * -3.125f);             // 1/(2*0.4^2)
      if (d > 0.8f) wr = 0.1f;                   // CLAMP_FACTOR*SIGMA_R -> MIN_VAL
      float w = (z <= muv) ? wl : wr;
      acc += l * w;
    }
    __syncthreads(); // compute done before next prefetch overwrites this buffer
    cur ^= 1;
  }

  // Cross-lane wave reduction via WMMA: D = ones(16x4) x B(4x16) + 0.
  // b = {acc, 0}: rows 0/2 of B hold the 32 lane accumulators, rows 1/3 zero.
  // Every element of D is a column sum; 4 xor-shuffles finish the 16 columns.
  v2f a;    a.x = 1.f;  a.y = 1.f;
  v2f bm;   bm.x = acc; bm.y = 0.f;
  v8f c = {};
  c = __builtin_amdgcn_wmma_f32_16x16x4_f32(false, a, false, bm, (short)0, c, false, false);
  float wsum = c[0];
#pragma unroll
  for (int m = 8; m >= 1; m >>= 1) wsum += __shfl_xor(wsum, m);
  if ((tid & 31) == 0) s_wave[tid >> 5] = wsum;
  __syncthreads();
  if (tid == 0) {
    float s = 0.f;
#pragma unroll
    for (int wv = 0; wv < NTHR / 32; ++wv) s += s_wave[wv];
    partial[blockIdx.x] = s; // every block writes (idle blocks write 0)
  }
}

// Deterministic fixed-order final reduction, then mean.
__global__ void gcel_final_kernel(const float* __restrict__ partial, int nblocks,
                                  int n, float* __restrict__ out) {
  __shared__ float s[NTHR];
  int tid = threadIdx.x;
  float a = 0.f;
  for (int j = tid; j < nblocks; j += NTHR) a += partial[j];
  s[tid] = a;
  __syncthreads();
  for (int m = NTHR / 2; m >= 1; m >>= 1) {
    if (tid < m) s[tid] += s[tid + m];
    __syncthreads();
  }
  if (tid == 0) out[0] = s[0] / (float)n;
}

extern "C" void kernel_launch(void* const* d_in, const int* in_sizes, int n_in,
                              void* d_out, int out_size, void* d_ws, size_t ws_size,
                              hipStream_t stream) {
  const float* pred    = (const float*)d_in[0]; // (N, 2)
  const float* coord   = (const float*)d_in[1]; // (N, 3)
  const int*   segment = (const int*)d_in[2];   // (N,)
  const int*   offset  = (const int*)d_in[3];   // (B,)
  int n = in_sizes[0] / 2;
  int B = in_sizes[3];
  if (B > MAXB) B = MAXB;

  int*   st      = (int*)d_ws;                                              // 4*MAXB ints
  float* mu      = (float*)((char*)d_ws + 4 * MAXB * sizeof(int));          // MAXB floats
  float* partial = (float*)((char*)d_ws + 4 * MAXB * sizeof(int)
                                        + MAXB * sizeof(float));            // NBLK_LOSS floats

  hipLaunchKernelGGL(gcel_init_kernel,  dim3(1),         dim3(128),  0, stream, st);
  hipLaunchKernelGGL(gcel_stats_kernel, dim3(NBLK_STAT), dim3(NTHR), 0, stream,
                     coord, segment, offset, n, B, st);
  hipLaunchKernelGGL(gcel_mu_kernel,    dim3(1),         dim3(64),   0, stream, st, mu, B);
  hipLaunchKernelGGL(gcel_loss_kernel,  dim3(NBLK_LOSS), dim3(NTHR), 0, stream,
                     pred, coord, segment, offset, mu, n, B, partial);
  hipLaunchKernelGGL(gcel_final_kernel, dim3(1),         dim3(NTHR), 0, stream,
                     partial, NBLK_LOSS, n, (float*)d_out);
}